// SAM_GRUCell_403726926425
// MI455X (gfx1250) — compile-verified
//
#include <hip/hip_runtime.h>
#include <math.h>

typedef __attribute__((ext_vector_type(2))) float v2f;
typedef __attribute__((ext_vector_type(8))) float v8f;

#define BSZ   8192
#define Hdim  128
#define Fdim  128
#define GN    1106          // N = 1100 + 3*W, W=2
#define NCELL (GN * GN)
#define WPAD  2

__device__ __forceinline__ float sigmoidf_(float x) { return 1.0f / (1.0f + expf(-x)); }

// K=128 fp32 WMMA dot: A row (16xK striped per ISA layout) x B rows (W^T)
// aBase/bBase already offset by the lane's (l>>4) float2 sub-offset.
__device__ __forceinline__ v8f wmma_k128(const float2* __restrict__ aBase,
                                         const float2* __restrict__ bBase,
                                         v8f acc)
{
#pragma unroll
    for (int i = 0; i < 32; ++i) {          // k = 4*i
        float2 af = aBase[2 * i];
        float2 bf = bBase[2 * i];
        v2f a;  a.x = af.x;  a.y = af.y;
        v2f bm; bm.x = bf.x; bm.y = bf.y;
        acc = __builtin_amdgcn_wmma_f32_16x16x4_f32(
            false, a, false, bm, (short)0, acc, false, false);
    }
    return acc;
}

// ---------------------------------------------------------------------------
// owner init / claim (deterministic duplicate-scatter resolution: max b wins)
// ---------------------------------------------------------------------------
__global__ void owner_init_kernel(int* __restrict__ owner) {
    int i = blockIdx.x * blockDim.x + threadIdx.x;
    if (i < NCELL) owner[i] = -1;
}

__global__ void owner_claim_kernel(const int* __restrict__ grid_idx, int* __restrict__ owner) {
    int b = blockIdx.x * blockDim.x + threadIdx.x;
    if (b >= BSZ) return;
    int gx = grid_idx[2 * b + 0] + WPAD;
    int gy = grid_idx[2 * b + 1] + WPAD;
    atomicMax(&owner[gx * GN + gy], b);
}

// ---------------------------------------------------------------------------
// Gates: per block = 16 batch rows.
//  combined[m][n] = x(m,:).w_ih(n,:) + hx(m,:).w_hh(n,:)   n in 0..511
//  hn[m][h]       = hx(m,:).w_hh(256+h,:)
// then fused elementwise gate math -> q, updategate, spatialgate (workspace)
// ---------------------------------------------------------------------------
__global__ __launch_bounds__(256) void gates_kernel(
    const float* __restrict__ xin, const float* __restrict__ hx,
    const float* __restrict__ w_ih, const float* __restrict__ w_hh,
    const float* __restrict__ b_ih, const float* __restrict__ b_hh,
    float* __restrict__ q_ws, float* __restrict__ ug_ws, float* __restrict__ sg_ws)
{
    __shared__ float sG[16 * 512];   // combined gates (no bias)
    __shared__ float sHn[16 * 128];  // hx-only n-gate   (no bias)

    const int tid = threadIdx.x;
    const int wv  = tid >> 5;        // wave id 0..7
    const int l   = tid & 31;        // lane
    const int lr  = l & 15;          // row/col within 16
    const int lh  = l >> 4;          // K half-select {0,1} (float2 units)
    const int row0 = blockIdx.x * 16;

    const float2* xrow = reinterpret_cast<const float2*>(xin + (size_t)(row0 + lr) * Fdim) + lh;
    const float2* hrow = reinterpret_cast<const float2*>(hx  + (size_t)(row0 + lr) * Hdim) + lh;

    // ---- combined GEMM: two K=128 passes, 4 col-tiles per wave ----
    for (int t = 0; t < 4; ++t) {
        const int col0 = (wv * 4 + t) * 16;
        const float2* wi = reinterpret_cast<const float2*>(w_ih + (size_t)(col0 + lr) * Fdim) + lh;
        const float2* wh = reinterpret_cast<const float2*>(w_hh + (size_t)(col0 + lr) * Hdim) + lh;
        v8f acc = {};
        acc = wmma_k128(xrow, wi, acc);   // x  . w_ih
        acc = wmma_k128(hrow, wh, acc);   // hx . w_hh
#pragma unroll
        for (int v = 0; v < 8; ++v)
            sG[(v + 8 * lh) * 512 + col0 + lr] = acc[v];
    }

    // ---- h_n tile: wave wv handles gate cols 256 + wv*16, K=128 ----
    {
        const int gcol = 256 + wv * 16;
        const float2* wh = reinterpret_cast<const float2*>(w_hh + (size_t)(gcol + lr) * Hdim) + lh;
        v8f acc = {};
        acc = wmma_k128(hrow, wh, acc);
#pragma unroll
        for (int v = 0; v < 8; ++v)
            sHn[(v + 8 * lh) * 128 + wv * 16 + lr] = acc[v];
    }
    __syncthreads();

    // ---- fused elementwise gate math (16*128 = 2048 elems, 8/thread) ----
    for (int i = 0; i < 8; ++i) {
        int idx = tid + i * 256;
        int m = idx >> 7;
        int h = idx & 127;
        int b = row0 + m;
        float r  = sigmoidf_(sG[m * 512 + h]       + b_ih[h]       + b_hh[h]);
        float u  = sigmoidf_(sG[m * 512 + 128 + h] + b_ih[128 + h] + b_hh[128 + h]);
        float s  = sigmoidf_(sG[m * 512 + 384 + h] + b_ih[384 + h] + b_hh[384 + h]);
        float hn = sHn[m * 128 + h] + b_hh[256 + h];
        float in_= sG[m * 512 + 256 + h] - sHn[m * 128 + h] + b_ih[256 + h];
        float q  = tanhf(in_ + r * hn);
        q_ws [(size_t)b * Hdim + h] = q;
        ug_ws[(size_t)b * Hdim + h] = u;
        sg_ws[(size_t)b * Hdim + h] = s;
    }
}

// ---------------------------------------------------------------------------
// Attention + output GEMM + state update + memory scatter.
// Block = 512 threads (16 waves) = 16 batch rows; one wave per row for the
// 5x5 neighborhood attention, then 8 WMMA waves for combined @ w_out^T.
// ---------------------------------------------------------------------------
__global__ __launch_bounds__(512) void attn_kernel(
    const float* __restrict__ hx, const int* __restrict__ grid_idx,
    const float* __restrict__ memory, const float* __restrict__ w_out,
    const float* __restrict__ b_out,
    const float* __restrict__ q_ws, const float* __restrict__ ug_ws,
    const float* __restrict__ sg_ws, const int* __restrict__ owner,
    float* __restrict__ hyy_out, float* __restrict__ new_memory)
{
    __shared__ float sComb[16 * 256]; // [mix | q] per row
    __shared__ float sOut[16 * 128];  // pre-activation attention output

    const int tid = threadIdx.x;
    const int wv  = tid >> 5;        // 0..15 : row within tile
    const int l   = tid & 31;        // lane : owns h = 4*l .. 4*l+3
    const int row0 = blockIdx.x * 16;
    const int bb  = row0 + wv;

    const int gx = grid_idx[2 * bb + 0] + WPAD;
    const int gy = grid_idx[2 * bb + 1] + WPAD;

    const float4 q4 = reinterpret_cast<const float4*>(q_ws + (size_t)bb * Hdim)[l];

    // single base pointer; the 25 neighbor offsets are compile-time constants
    // that fold into the load's signed-24-bit instruction offset.
    const float* cbase = memory + ((size_t)gx * GN + gy) * Hdim + 4 * l;

    // ---- scores: 25 neighbor dot products, butterfly-reduced over wave ----
    float sc[25];
#pragma unroll
    for (int n = 0; n < 25; ++n) {
        const int dx = n / 5 - 2, dy = n % 5 - 2;
        float4 c4 = *reinterpret_cast<const float4*>(cbase + (dx * GN + dy) * Hdim);
        float p = q4.x * c4.x + q4.y * c4.y + q4.z * c4.z + q4.w * c4.w;
#pragma unroll
        for (int o = 16; o >= 1; o >>= 1) p += __shfl_xor(p, o, 32);
        sc[n] = p;
    }

    // ---- softmax in registers (replicated across lanes) ----
    float mx = sc[0];
#pragma unroll
    for (int n = 1; n < 25; ++n) mx = fmaxf(mx, sc[n]);
    float sum = 0.0f;
#pragma unroll
    for (int n = 0; n < 25; ++n) { sc[n] = expf(sc[n] - mx); sum += sc[n]; }
    const float inv = 1.0f / sum;

    // ---- mix = sum_l attn_l * cs_l  (re-load, L1/L2 resident) ----
    float mixx = 0.f, mixy = 0.f, mixz = 0.f, mixw = 0.f;
#pragma unroll
    for (int n = 0; n < 25; ++n) {
        const int dx = n / 5 - 2, dy = n % 5 - 2;
        float4 c4 = *reinterpret_cast<const float4*>(cbase + (dx * GN + dy) * Hdim);
        float a = sc[n] * inv;
        mixx += a * c4.x; mixy += a * c4.y; mixz += a * c4.z; mixw += a * c4.w;
    }
    sComb[wv * 256 + 4 * l + 0] = mixx;
    sComb[wv * 256 + 4 * l + 1] = mixy;
    sComb[wv * 256 + 4 * l + 2] = mixz;
    sComb[wv * 256 + 4 * l + 3] = mixw;
    sComb[wv * 256 + 128 + 4 * l + 0] = q4.x;
    sComb[wv * 256 + 128 + 4 * l + 1] = q4.y;
    sComb[wv * 256 + 128 + 4 * l + 2] = q4.z;
    sComb[wv * 256 + 128 + 4 * l + 3] = q4.w;
    __syncthreads();

    // ---- out16 = combined(16x256) @ w_out^T (256x128): 8 WMMA waves ----
    if (wv < 8) {
        const int lr = l & 15;
        const int lh = l >> 4;
        const int n0 = wv * 16;
        const float2* aB = reinterpret_cast<const float2*>(&sComb[lr * 256]) + lh;
        const float2* wo = reinterpret_cast<const float2*>(w_out + (size_t)(n0 + lr) * 256) + lh;
        v8f acc = {};
        acc = wmma_k128(aB, wo, acc);             // K = 0..127   (mix half)
        acc = wmma_k128(aB + 64, wo + 64, acc);   // K = 128..255 (q half)
#pragma unroll
        for (int v = 0; v < 8; ++v)
            sOut[(v + 8 * lh) * 128 + n0 + lr] = acc[v];
    }
    __syncthreads();

    // ---- fused tail: hyy + deterministic memory-cell update ----
    for (int i = 0; i < 4; ++i) {
        int idx = tid + i * 512;
        int m = idx >> 7;
        int h = idx & 127;
        int b2 = row0 + m;
        float at = tanhf(sOut[m * 128 + h] + b_out[h]);
        float q  = sComb[m * 256 + 128 + h];
        float sg = sg_ws[(size_t)b2 * Hdim + h];
        float ug = ug_ws[(size_t)b2 * Hdim + h];
        float hv = hx[(size_t)b2 * Hdim + h];
        float curr = q + sg * at;
        float hy   = curr + ug * (hv - curr);
        hyy_out[(size_t)b2 * Hdim + h] = hy;

        int gx2 = grid_idx[2 * b2 + 0] + WPAD;
        int gy2 = grid_idx[2 * b2 + 1] + WPAD;
        size_t cell = (size_t)gx2 * GN + gy2;
        if (owner[cell] == b2) {
            float mc = memory[cell * Hdim + h];
            new_memory[cell * Hdim + h] = sg * mc + (1.0f - sg) * hy;
        }
    }
}

// ---------------------------------------------------------------------------
extern "C" void kernel_launch(void* const* d_in, const int* in_sizes, int n_in,
                              void* d_out, int out_size, void* d_ws, size_t ws_size,
                              hipStream_t stream) {
    const float* input_feat = (const float*)d_in[0];
    const float* hx         = (const float*)d_in[1];
    const int*   grid_idx   = (const int*)  d_in[2];
    const float* memory     = (const float*)d_in[3];
    const float* w_ih       = (const float*)d_in[4];
    const float* w_hh       = (const float*)d_in[5];
    const float* b_ih       = (const float*)d_in[6];
    const float* b_hh       = (const float*)d_in[7];
    const float* w_out      = (const float*)d_in[8];
    const float* b_out      = (const float*)d_in[9];

    float* hyy        = (float*)d_out;
    float* new_memory = hyy + (size_t)BSZ * Hdim;

    float* q_ws  = (float*)d_ws;
    float* ug_ws = q_ws  + (size_t)BSZ * Hdim;
    float* sg_ws = ug_ws + (size_t)BSZ * Hdim;
    int*   owner = (int*)(sg_ws + (size_t)BSZ * Hdim);

    // bulk move of the 626 MB memory tensor at copy-engine bandwidth
    hipMemcpyAsync(new_memory, memory, (size_t)NCELL * Hdim * sizeof(float),
                   hipMemcpyDeviceToDevice, stream);

    owner_init_kernel <<<(NCELL + 255) / 256, 256, 0, stream>>>(owner);
    owner_claim_kernel<<<(BSZ   + 255) / 256, 256, 0, stream>>>(grid_idx, owner);

    gates_kernel<<<BSZ / 16, 256, 0, stream>>>(input_feat, hx, w_ih, w_hh,
                                               b_ih, b_hh, q_ws, ug_ws, sg_ws);

    attn_kernel<<<BSZ / 16, 512, 0, stream>>>(hx, grid_idx, memory, w_out, b_out,
                                              q_ws, ug_ws, sg_ws, owner,
                                              hyy, new_memory);
}